// GATClassifier_7645041787171
// MI455X (gfx1250) — compile-verified
//
#include <hip/hip_runtime.h>
#include <math.h>

typedef __attribute__((ext_vector_type(2))) float v2f;
typedef __attribute__((ext_vector_type(8))) float v8f;

#define N_SIDE 8192
#define N_TOT  16384
#define IN_DIM 128
#define D1     256     // H*HID
#define HID    64
#define NHEAD  4
#define NPG    128     // nodes per graph
#define NGRAPH 128     // 2*B_SIDE
#define BSIDE  64

static __device__ __forceinline__ v8f wmma_f32(v2f a, v2f b, v8f c) {
    return __builtin_amdgcn_wmma_f32_16x16x4_f32(false, a, false, b, (short)0, c,
                                                 false, false);
}

// -------- async global->LDS support (guarded; falls back to sync copy) ------
#ifndef USE_ASYNC_LDS
#  if defined(__has_builtin)
#    if __has_builtin(__builtin_amdgcn_global_load_async_to_lds_b128) && \
        __has_builtin(__builtin_amdgcn_s_wait_asynccnt)
#      define USE_ASYNC_LDS 1
#    else
#      define USE_ASYNC_LDS 0
#    endif
#  else
#    define USE_ASYNC_LDS 0
#  endif
#endif

#if USE_ASYNC_LDS
typedef int v4i_ __attribute__((vector_size(16)));                 // 16-byte payload
typedef __attribute__((address_space(1))) v4i_* as1v4p;            // global ptr
typedef __attribute__((address_space(3))) v4i_* as3v4p;            // LDS ptr
static __device__ __forceinline__ void async_copy16(const float* g, float* l) {
    // LDS offset is the low 32 bits of the generic pointer (aperture scheme)
    __builtin_amdgcn_global_load_async_to_lds_b128(
        (as1v4p)(uintptr_t)g, (as3v4p)(unsigned int)(uintptr_t)l, 0, 0);
}
#endif

// ---------------------------------------------------------------------------
// Kernel A: flash-attention style  O = softmax(Q @ K^T) @ K   (K doubles as V)
// Q,K: [8192,128] fp32.  Block: 256 threads (8 waves), 64 Q rows per block.
// Q fragment held in registers; K tile double-buffered in LDS with async
// global->LDS loads overlapped with WMMA compute.
// ---------------------------------------------------------------------------
#define TM 64
#define TN 32
#define KS_LD 132
#define SS_LD 36
#define NTILES (N_SIDE / TN)

__global__ __launch_bounds__(256) void attn_kernel(const float* __restrict__ Q,
                                                   const float* __restrict__ KV,
                                                   float* __restrict__ O) {
    __shared__ float smem[TM * KS_LD];          // Q staging, then 2x K-tile buffers
    __shared__ float Ss[TM][SS_LD];
    __shared__ float mrow[TM], lrow[TM], rscale[TM];

    const int tid  = threadIdx.x;
    const int lane = tid & 31, wv = tid >> 5;
    const int lr = lane & 15, lh = lane >> 4;
    const int r0 = blockIdx.x * TM;
    const int rt = wv >> 1;       // score row-tile (0..3)
    const int ctS = wv & 1;       // score col-tile (0..1)

    // stage 64x128 Q tile through LDS (coalesced), pull per-lane A frags to regs
    for (int i = tid; i < TM * (IN_DIM / 4); i += 256) {
        int row = i >> 5, c4 = (i & 31) << 2;
        float4 v = *(const float4*)(Q + (size_t)(r0 + row) * IN_DIM + c4);
        float* p = &smem[row * KS_LD + c4];
        p[0] = v.x; p[1] = v.y; p[2] = v.z; p[3] = v.w;
    }
    if (tid < TM) { mrow[tid] = -1e30f; lrow[tid] = 0.f; }
    __syncthreads();

    v2f qa[32];                                 // loop-invariant A fragment
    {
        const float* qrow = &smem[(rt * 16 + lr) * KS_LD + lh * 2];
        #pragma unroll
        for (int p = 0; p < 32; p++)
            qa[p] = *(const v2f*)(qrow + 4 * p);
    }
    __syncthreads();

    v8f oacc[4];
    #pragma unroll
    for (int i = 0; i < 4; i++) oacc[i] = (v8f){};

    float* const Kb0 = smem;
    float* const Kb1 = smem + TN * KS_LD;

    // prologue: fetch K tile 0 into buffer 0
    for (int i = tid; i < TN * (IN_DIM / 4); i += 256) {
        int row = i >> 5, c4 = (i & 31) << 2;
        const float* g = KV + (size_t)row * IN_DIM + c4;
        float* l = Kb0 + row * KS_LD + c4;
#if USE_ASYNC_LDS
        async_copy16(g, l);
#else
        float4 v = *(const float4*)g;
        l[0] = v.x; l[1] = v.y; l[2] = v.z; l[3] = v.w;
#endif
    }

    for (int jt = 0; jt < NTILES; jt++) {
        float* const Kc = (jt & 1) ? Kb1 : Kb0;
        float* const Kn = (jt & 1) ? Kb0 : Kb1;
#if USE_ASYNC_LDS
        __builtin_amdgcn_s_wait_asynccnt(0);
#endif
        __syncthreads();                         // tile jt resident in Kc

        // issue prefetch of tile jt+1 into the other buffer (overlaps compute)
        if (jt + 1 < NTILES) {
            const int j0 = (jt + 1) * TN;
            for (int i = tid; i < TN * (IN_DIM / 4); i += 256) {
                int row = i >> 5, c4 = (i & 31) << 2;
                const float* g = KV + (size_t)(j0 + row) * IN_DIM + c4;
                float* l = Kn + row * KS_LD + c4;
#if USE_ASYNC_LDS
                async_copy16(g, l);
#else
                float4 v = *(const float4*)g;
                l[0] = v.x; l[1] = v.y; l[2] = v.z; l[3] = v.w;
#endif
            }
        }

        // S tile = Q_tile @ K_tile^T  (one 16x16 per wave, 32 k-steps of 4)
        v8f sc = (v8f){};
        {
            const float* brow = Kc + (ctS * 16 + lr) * KS_LD + lh * 2;
            #pragma unroll
            for (int p = 0; p < 32; p++) {
                v2f b = *(const v2f*)(brow + 4 * p);
                sc = wmma_f32(qa[p], b, sc);
            }
        }
        #pragma unroll
        for (int r = 0; r < 8; r++)
            Ss[rt * 16 + r + lh * 8][ctS * 16 + lr] = sc[r];
        __syncthreads();

        // online softmax stats: thread t handles row t (t < 64)
        if (tid < TM) {
            float m_old = mrow[tid], mt = m_old;
            for (int c = 0; c < TN; c++) mt = fmaxf(mt, Ss[tid][c]);
            float scale = __expf(m_old - mt);
            float ls = 0.f;
            for (int c = 0; c < TN; c++) {
                float p = __expf(Ss[tid][c] - mt);
                Ss[tid][c] = p;
                ls += p;
            }
            mrow[tid] = mt;
            lrow[tid] = lrow[tid] * scale + ls;
            rscale[tid] = scale;
        }
        __syncthreads();

        // rescale accumulators, then O += P @ V  (wave owns 4 col-tiles)
        #pragma unroll
        for (int i = 0; i < 4; i++)
            #pragma unroll
            for (int r = 0; r < 8; r++)
                oacc[i][r] *= rscale[rt * 16 + r + lh * 8];

        #pragma unroll
        for (int i = 0; i < 4; i++) {
            const int c0 = ((wv & 1) + 2 * i) * 16;
            v8f acc = oacc[i];
            for (int k = 0; k < TN; k += 4) {
                v2f a = *(const v2f*)(&Ss[rt * 16 + lr][k + lh * 2]);
                v2f b;
                b.x = Kc[(k + lh * 2) * KS_LD + c0 + lr];
                b.y = Kc[(k + lh * 2 + 1) * KS_LD + c0 + lr];
                acc = wmma_f32(a, b, acc);
            }
            oacc[i] = acc;
        }
    }
    __syncthreads();
    // normalize by l and store
    #pragma unroll
    for (int i = 0; i < 4; i++) {
        const int c0 = ((wv & 1) + 2 * i) * 16 + lr;
        #pragma unroll
        for (int r = 0; r < 8; r++) {
            int row = rt * 16 + r + lh * 8;
            O[(size_t)(r0 + row) * IN_DIM + c0] = oacc[i][r] / lrow[row];
        }
    }
}

// ---------------------------------------------------------------------------
// Kernel B: h = [[nbd1,x1a,nbd1-x1a,nbd1*x1a];[nbd2,x2a,nbd2-x2a,nbd2*x2a]]
// ---------------------------------------------------------------------------
__global__ void buildh_kernel(const float* __restrict__ nbd1,
                              const float* __restrict__ nbd2,
                              const float* __restrict__ x1a,
                              const float* __restrict__ x2a,
                              float* __restrict__ h) {
    size_t idx = (size_t)blockIdx.x * blockDim.x + threadIdx.x;
    if (idx >= (size_t)N_TOT * 512) return;
    int n = (int)(idx >> 9), c = (int)(idx & 511);
    int seg = c >> 7, cc = c & 127;
    const float *xb, *xa; int bn;
    if (n < N_SIDE) { xb = nbd1; xa = x1a; bn = n; }
    else            { xb = nbd2; xa = x2a; bn = n - N_SIDE; }
    float x = xb[(size_t)bn * IN_DIM + cc];
    float a = xa[(size_t)bn * IN_DIM + cc];
    float v;
    switch (seg) { case 0: v = x; break; case 1: v = a; break;
                   case 2: v = x - a; break; default: v = x * a; }
    h[idx] = v;
}

// ---------------------------------------------------------------------------
// Kernel C: C[M,N] = A[M,K] @ B[N,K]^T  via WMMA f32. 64x64 tile per block.
// ---------------------------------------------------------------------------
__global__ __launch_bounds__(256) void gemm_wmma(const float* __restrict__ A,
                                                 const float* __restrict__ B,
                                                 float* __restrict__ C,
                                                 int M, int N, int K) {
    __shared__ float As[64][68];
    __shared__ float Bs[64][68];
    const int tid = threadIdx.x, lane = tid & 31, wv = tid >> 5;
    const int lr = lane & 15, lh = lane >> 4;
    const int m0 = blockIdx.x * 64, n0 = blockIdx.y * 64;
    const int rt = wv >> 1;
    v8f acc[2]; acc[0] = (v8f){}; acc[1] = (v8f){};

    for (int k0 = 0; k0 < K; k0 += 64) {
        __syncthreads();
        for (int i = tid; i < 64 * 16; i += 256) {
            int row = i >> 4, c4 = (i & 15) << 2;
            float4 va = *(const float4*)(A + (size_t)(m0 + row) * K + k0 + c4);
            As[row][c4] = va.x; As[row][c4 + 1] = va.y;
            As[row][c4 + 2] = va.z; As[row][c4 + 3] = va.w;
            float4 vb = *(const float4*)(B + (size_t)(n0 + row) * K + k0 + c4);
            Bs[row][c4] = vb.x; Bs[row][c4 + 1] = vb.y;
            Bs[row][c4 + 2] = vb.z; Bs[row][c4 + 3] = vb.w;
        }
        __syncthreads();
        const int arow = rt * 16 + lr;
        #pragma unroll
        for (int t = 0; t < 2; t++) {
            const int bcol = ((wv & 1) + 2 * t) * 16 + lr;
            v8f a_ = acc[t];
            for (int k = 0; k < 64; k += 4) {
                v2f a = *(const v2f*)(&As[arow][k + lh * 2]);
                v2f b = *(const v2f*)(&Bs[bcol][k + lh * 2]);
                a_ = wmma_f32(a, b, a_);
            }
            acc[t] = a_;
        }
    }
    #pragma unroll
    for (int t = 0; t < 2; t++) {
        int col = n0 + ((wv & 1) + 2 * t) * 16 + lr;
        #pragma unroll
        for (int r = 0; r < 8; r++) {
            int row = m0 + rt * 16 + r + lh * 8;
            C[(size_t)row * N + col] = acc[t][r];
        }
    }
}

// ---------------------------------------------------------------------------
// Kernel D: per-(node,head) attention scalars a1,a2
// ---------------------------------------------------------------------------
__global__ void scores_kernel(const float* __restrict__ ft,
                              const float* __restrict__ al,
                              const float* __restrict__ ar,
                              float* __restrict__ aL, float* __restrict__ aR) {
    int idx = blockIdx.x * blockDim.x + threadIdx.x;  // N_TOT*4
    if (idx >= N_TOT * NHEAD) return;
    int node = idx >> 2, hh = idx & 3;
    const float* f = ft + (size_t)node * D1 + hh * HID;
    const float* l = al + hh * HID;
    const float* r = ar + hh * HID;
    float s1 = 0.f, s2 = 0.f;
    for (int d = 0; d < HID; d++) { float v = f[d]; s1 += v * l[d]; s2 += v * r[d]; }
    aL[idx] = s1; aR[idx] = s2;
}

// ---------------------------------------------------------------------------
// Kernel E: per-graph GAT edge softmax + aggregation + ELU (one block/graph)
// ---------------------------------------------------------------------------
__global__ __launch_bounds__(256) void gat_agg_kernel(const float* __restrict__ ft,
                                                      const float* __restrict__ aL,
                                                      const float* __restrict__ aR,
                                                      const int* __restrict__ src,
                                                      const int* __restrict__ dst,
                                                      float* __restrict__ out) {
    __shared__ float a1s[NPG][4], a2s[NPG][4], ea[NPG][4];
    __shared__ float mm[NPG][4], ssum[NPG][4];
    __shared__ int dloc[NPG], sloc[NPG];
    __shared__ float acc[NPG][HID];   // 32 KB, reused per head
    const int g = blockIdx.x, base = g * NPG, tid = threadIdx.x;

    for (int i = tid; i < NPG * 4; i += 256) {
        a1s[i >> 2][i & 3] = aL[base * 4 + i];
        a2s[i >> 2][i & 3] = aR[base * 4 + i];
    }
    if (tid < NPG) {
        sloc[tid] = src[base + tid] - base;
        dloc[tid] = dst[base + tid] - base;
    }
    __syncthreads();
    for (int i = tid; i < NPG * 4; i += 256) {
        int e = i >> 2, hh = i & 3;
        float v = a1s[sloc[e]][hh] + a2s[dloc[e]][hh];
        ea[e][hh] = v > 0.f ? v : 0.2f * v;   // LeakyReLU
    }
    __syncthreads();
    if (tid < NPG) {   // stable edge-softmax stats per dst node
        for (int hh = 0; hh < 4; hh++) {
            float m = -1e30f;
            for (int e = 0; e < NPG; e++)
                if (dloc[e] == tid) m = fmaxf(m, ea[e][hh]);
            float s = 0.f;
            for (int e = 0; e < NPG; e++)
                if (dloc[e] == tid) s += __expf(ea[e][hh] - m);
            mm[tid][hh] = m; ssum[tid][hh] = s;
        }
    }
    __syncthreads();
    for (int i = tid; i < NPG * 4; i += 256) {
        int e = i >> 2, hh = i & 3, d = dloc[e];
        ea[e][hh] = __expf(ea[e][hh] - mm[d][hh]) / ssum[d][hh];
    }
    // aggregate per head through LDS (ds_add_f32 atomics), apply ELU
    for (int hh = 0; hh < 4; hh++) {
        __syncthreads();
        for (int i = tid; i < NPG * HID; i += 256) acc[i >> 6][i & 63] = 0.f;
        __syncthreads();
        for (int i = tid; i < NPG * HID; i += 256) {
            int e = i >> 6, f = i & 63;
            atomicAdd(&acc[dloc[e]][f],
                      ft[(size_t)(base + sloc[e]) * D1 + hh * HID + f] * ea[e][hh]);
        }
        __syncthreads();
        for (int i = tid; i < NPG * HID; i += 256) {
            int d = i >> 6, f = i & 63;
            float v = acc[d][f];
            out[(size_t)(base + d) * D1 + hh * HID + f] = v > 0.f ? v : expm1f(v);
        }
    }
}

// ---------------------------------------------------------------------------
// Readout + MLP tail
// ---------------------------------------------------------------------------
__global__ void pool_kernel(const float* __restrict__ h2,
                            const float* __restrict__ ebd,
                            float* __restrict__ hg) {
    int g = blockIdx.x, f = threadIdx.x;   // 320 threads
    float s = 0.f;
    if (f < D1) {
        for (int n = 0; n < NPG; n++) s += h2[(size_t)(g * NPG + n) * D1 + f];
    } else {
        int f2 = f - D1;
        for (int n = 0; n < NPG; n++) s += ebd[(size_t)(g * NPG + n) * HID + f2];
    }
    hg[g * 320 + f] = s * (1.f / 128.f);
}

__global__ void bn1_kernel(const float* __restrict__ hg,
                           const float* __restrict__ gamma,
                           const float* __restrict__ beta,
                           float* __restrict__ xbn) {
    int c = threadIdx.x;   // 640 columns
    float mu = 0.f, m2 = 0.f;
    for (int b = 0; b < BSIDE; b++) {
        float v = (c < 320) ? hg[b * 320 + c] : hg[(b + BSIDE) * 320 + (c - 320)];
        mu += v; m2 += v * v;
    }
    mu *= (1.f / 64.f); m2 *= (1.f / 64.f);
    float inv = rsqrtf(m2 - mu * mu + 1e-5f);
    float gsc = gamma[c] * inv, bb = beta[c];
    for (int b = 0; b < BSIDE; b++) {
        float v = (c < 320) ? hg[b * 320 + c] : hg[(b + BSIDE) * 320 + (c - 320)];
        xbn[b * 640 + c] = gsc * (v - mu) + bb;
    }
}

__global__ void lin1_kernel(const float* __restrict__ x,
                            const float* __restrict__ w,
                            const float* __restrict__ bias,
                            float* __restrict__ y) {
    int idx = blockIdx.x * blockDim.x + threadIdx.x;   // 64*64
    if (idx >= BSIDE * HID) return;
    int b = idx >> 6, o = idx & 63;
    float s = bias[o];
    const float* xr = x + b * 640;
    const float* wr = w + o * 640;
    for (int c = 0; c < 640; c++) s += xr[c] * wr[c];
    y[idx] = fmaxf(s, 0.f);
}

__global__ void bn2_kernel(const float* __restrict__ y1,
                           const float* __restrict__ gamma,
                           const float* __restrict__ beta,
                           float* __restrict__ y2) {
    int o = threadIdx.x;   // 64 columns
    float mu = 0.f, m2 = 0.f;
    for (int b = 0; b < BSIDE; b++) { float v = y1[b * HID + o]; mu += v; m2 += v * v; }
    mu *= (1.f / 64.f); m2 *= (1.f / 64.f);
    float inv = rsqrtf(m2 - mu * mu + 1e-5f);
    float gsc = gamma[o] * inv, bb = beta[o];
    for (int b = 0; b < BSIDE; b++)
        y2[b * HID + o] = gsc * (y1[b * HID + o] - mu) + bb;
}

__global__ void lin2_kernel(const float* __restrict__ y,
                            const float* __restrict__ w,
                            const float* __restrict__ bias,
                            float* __restrict__ out) {
    int idx = threadIdx.x;
    if (idx >= BSIDE * 3) return;
    int b = idx / 3, k = idx % 3;
    float s = bias[k];
    for (int o = 0; o < HID; o++) s += y[b * HID + o] * w[k * HID + o];
    out[idx] = s;
}

// ---------------------------------------------------------------------------
extern "C" void kernel_launch(void* const* d_in, const int* in_sizes, int n_in,
                              void* d_out, int out_size, void* d_ws, size_t ws_size,
                              hipStream_t stream) {
    const float* nbd1 = (const float*)d_in[0];
    const float* nbd2 = (const float*)d_in[1];
    const float* ebd  = (const float*)d_in[2];
    const float* W1   = (const float*)d_in[3];
    const float* al1  = (const float*)d_in[4];
    const float* ar1  = (const float*)d_in[5];
    const float* W2   = (const float*)d_in[6];
    const float* al2  = (const float*)d_in[7];
    const float* ar2  = (const float*)d_in[8];
    const float* bn1g = (const float*)d_in[9];
    const float* bn1b = (const float*)d_in[10];
    const float* l1w  = (const float*)d_in[11];
    const float* l1b  = (const float*)d_in[12];
    const float* bn2g = (const float*)d_in[13];
    const float* bn2b = (const float*)d_in[14];
    const float* l2w  = (const float*)d_in[15];
    const float* l2b  = (const float*)d_in[16];
    const int*   src  = (const int*)d_in[17];
    const int*   dst  = (const int*)d_in[18];
    float* out = (float*)d_out;

    float* ws  = (float*)d_ws;
    float* x1a = ws;                                   // 8192*128
    float* x2a = x1a + (size_t)N_SIDE * IN_DIM;        // 8192*128
    float* h   = x2a + (size_t)N_SIDE * IN_DIM;        // 16384*512
    float* ft  = h   + (size_t)N_TOT * 512;            // 16384*256 (both layers)
    float* aLb = ft  + (size_t)N_TOT * D1;             // 16384*4
    float* aRb = aLb + (size_t)N_TOT * NHEAD;          // 16384*4
    float* h1  = aRb + (size_t)N_TOT * NHEAD;          // 16384*256
    float* h2  = h1  + (size_t)N_TOT * D1;             // 16384*256
    float* hg  = h2  + (size_t)N_TOT * D1;             // 128*320
    float* xbn = hg  + (size_t)NGRAPH * 320;           // 64*640
    float* y1  = xbn + (size_t)BSIDE * 640;            // 64*64
    float* y2  = y1  + (size_t)BSIDE * HID;            // 64*64

    // Node alignment (flash attention both directions)
    attn_kernel<<<N_SIDE / TM, 256, 0, stream>>>(nbd1, nbd2, x1a);
    attn_kernel<<<N_SIDE / TM, 256, 0, stream>>>(nbd2, nbd1, x2a);

    // h = concat features [16384, 512]
    buildh_kernel<<<(int)(((size_t)N_TOT * 512) / 256), 256, 0, stream>>>(
        nbd1, nbd2, x1a, x2a, h);

    // GAT layer 1
    gemm_wmma<<<dim3(N_TOT / 64, D1 / 64), 256, 0, stream>>>(h, W1, ft, N_TOT, D1, 512);
    scores_kernel<<<(N_TOT * NHEAD + 255) / 256, 256, 0, stream>>>(ft, al1, ar1, aLb, aRb);
    gat_agg_kernel<<<NGRAPH, 256, 0, stream>>>(ft, aLb, aRb, src, dst, h1);

    // GAT layer 2
    gemm_wmma<<<dim3(N_TOT / 64, D1 / 64), 256, 0, stream>>>(h1, W2, ft, N_TOT, D1, D1);
    scores_kernel<<<(N_TOT * NHEAD + 255) / 256, 256, 0, stream>>>(ft, al2, ar2, aLb, aRb);
    gat_agg_kernel<<<NGRAPH, 256, 0, stream>>>(ft, aLb, aRb, src, dst, h2);

    // Readout + MLP
    pool_kernel<<<NGRAPH, 320, 0, stream>>>(h2, ebd, hg);
    bn1_kernel<<<1, 640, 0, stream>>>(hg, bn1g, bn1b, xbn);
    lin1_kernel<<<16, 256, 0, stream>>>(xbn, l1w, l1b, y1);
    bn2_kernel<<<1, 64, 0, stream>>>(y1, bn2g, bn2b, y2);
    lin2_kernel<<<1, 192, 0, stream>>>(y2, l2w, l2b, out);
}